// RandomChiNoise_40235253629142
// MI455X (gfx1250) — compile-verified
//
#include <hip/hip_runtime.h>

// out[b,c,i] = image[b,c,i] + sqrt(sigma[b,c]^2 * (eps[0,b,c,i]^2 + sum_{n>=2*nc[b]+2} eps[n,b,c,i]^2))
//
// Memory-bound streaming kernel for MI455X (gfx1250):
//  - 128-bit vector loads/stores (global_load_b128 / global_store_b128)
//  - non-temporal cache hints (TH=NT): ~300MB one-shot stream > 192MB L2
//  - global_prefetch_b8 of the next eps plane (32MB stride between planes)
//  - per-batch mask is wave-uniform -> scalar branch, no dropped-plane traffic
//  - channel/batch come from grid.y/grid.z: no 64-bit division in the kernel

typedef float f4 __attribute__((ext_vector_type(4)));

__global__ __launch_bounds__(256) void chi_noise_add_v4(
    const float* __restrict__ image,
    const float* __restrict__ sigma,
    const float* __restrict__ eps,
    const int*   __restrict__ ncoils,
    float*       __restrict__ out,
    long long S4,     // float4s per (b,c) channel volume
    long long M4,     // float4s per batch (= C*S4)
    int C, int B, int K)
{
    const int c = blockIdx.y;
    const int b = blockIdx.z;
    const long long j4 = (long long)blockIdx.x * blockDim.x + threadIdx.x;
    if (j4 >= S4) return;
    const long long i4 = (long long)c * S4 + j4;   // float4 index within batch

    // eps plane p (1..K-1) kept iff p > 2*ncoils[b]+1  ->  p >= 2*ncoils[b]+2
    const int start = 2 * ncoils[b] + 2;           // uniform across the block
    const float sg = sigma[b * C + c];
    const float s2 = sg * sg;

    const f4* epsb = (const f4*)eps + (long long)b * M4;   // plane 0 of batch b
    const long long pstride = (long long)B * M4;           // float4s between planes

    // plane 0 always contributes
    f4 e0 = __builtin_nontemporal_load(epsb + i4);
    f4 acc = e0 * e0;

    // kept loop planes (uniform trip count per batch)
    for (int n = start; n < K; ++n) {
        const f4* p = epsb + (long long)n * pstride + i4;
        if (n + 1 < K) {
            // pull next 32MB-distant plane toward L2 while we consume this one
            __builtin_prefetch((const void*)(p + pstride), 0, 1);
        }
        f4 en = __builtin_nontemporal_load(p);
        acc += en * en;
    }

    f4 im = __builtin_nontemporal_load((const f4*)image + (long long)b * M4 + i4);
    f4 r;
    r.x = im.x + sqrtf(s2 * acc.x);
    r.y = im.y + sqrtf(s2 * acc.y);
    r.z = im.z + sqrtf(s2 * acc.z);
    r.w = im.w + sqrtf(s2 * acc.w);
    __builtin_nontemporal_store(r, (f4*)out + (long long)b * M4 + i4);
}

// Scalar fallback for shapes whose channel volume is not a multiple of 4.
__global__ __launch_bounds__(256) void chi_noise_add_scalar(
    const float* __restrict__ image,
    const float* __restrict__ sigma,
    const float* __restrict__ eps,
    const int*   __restrict__ ncoils,
    float*       __restrict__ out,
    long long S, long long M, int C, int B, int K)
{
    const int c = blockIdx.y;
    const int b = blockIdx.z;
    const long long j = (long long)blockIdx.x * blockDim.x + threadIdx.x;
    if (j >= S) return;
    const long long i = (long long)c * S + j;

    const int start = 2 * ncoils[b] + 2;
    const float sg = sigma[b * C + c];
    const float s2 = sg * sg;

    const float* epsb = eps + (long long)b * M;
    const long long pstride = (long long)B * M;

    float e0 = __builtin_nontemporal_load(epsb + i);
    float acc = e0 * e0;
    for (int n = start; n < K; ++n) {
        float en = __builtin_nontemporal_load(epsb + (long long)n * pstride + i);
        acc += en * en;
    }
    float im = __builtin_nontemporal_load(image + (long long)b * M + i);
    __builtin_nontemporal_store(im + sqrtf(s2 * acc), out + (long long)b * M + i);
}

extern "C" void kernel_launch(void* const* d_in, const int* in_sizes, int n_in,
                              void* d_out, int out_size, void* d_ws, size_t ws_size,
                              hipStream_t stream) {
    const float* image  = (const float*)d_in[0];
    const float* sigma  = (const float*)d_in[1];
    const float* eps    = (const float*)d_in[2];
    const int*   ncoils = (const int*)d_in[3];
    float*       out    = (float*)d_out;

    const int       B     = in_sizes[3];                 // per-batch coil counts
    const long long total = (long long)in_sizes[0];      // B*C*X*Y*Z
    const long long M     = total / B;                   // C*X*Y*Z per batch
    const int       C     = in_sizes[1] / B;             // sigma is [B,C]
    const long long S     = M / C;                       // X*Y*Z per channel
    const int       K     = (int)((long long)in_sizes[2] / total);  // eps planes

    const int threads = 256;
    if ((S & 3LL) == 0) {
        const long long S4 = S >> 2;
        const long long M4 = M >> 2;
        dim3 grid((unsigned)((S4 + threads - 1) / threads), (unsigned)C, (unsigned)B);
        chi_noise_add_v4<<<grid, dim3(threads), 0, stream>>>(
            image, sigma, eps, ncoils, out, S4, M4, C, B, K);
    } else {
        dim3 grid((unsigned)((S + threads - 1) / threads), (unsigned)C, (unsigned)B);
        chi_noise_add_scalar<<<grid, dim3(threads), 0, stream>>>(
            image, sigma, eps, ncoils, out, S, M, C, B, K);
    }
}